// REINFORCE_8358006358594
// MI455X (gfx1250) — compile-verified
//
#include <hip/hip_runtime.h>

// out = sum_{b,t} (t+1) * w^2 * log(sigmoid(logit)+eps) * log_prob  /  sum(w)
// (reverse-cumsum-then-sum collapses to the (t+1)-weighted sum).
// Pure HBM-bandwidth-bound streaming reduction: ~100.7 MB @ 23.3 TB/s ~= 4.3 us.

typedef __attribute__((ext_vector_type(2))) float v2f;
typedef __attribute__((ext_vector_type(8))) float v8f;

#define T_MASK   16383      // T = 16384, power of two
#define NBLOCKS  2048
#define NTHREADS 256
#define NWAVES   (NTHREADS / 32)
#define EPS_F    1e-7f

// Reduce one float across the 32 lanes of a wave using V_WMMA_F32_16X16X4_F32.
// A (16x4 f32) layout: lanes 0-15 hold A[m][0],A[m][1]; lanes 16-31 hold A[m][2],A[m][3].
// Put the partial in the first slot, 0 in the second; B = ones(4x16).
// D[m][n] = p_m + p_{m+16} for every n. D's 8 VGPRs at lane n<16 carry rows 0-7,
// at lane n>=16 rows 8-15 -> sum own 8 VGPRs, then xor-shuffle across the half-waves.
__device__ __forceinline__ float wave_sum_wmma(float x) {
  v2f a; a[0] = x;    a[1] = 0.0f;
  v2f b; b[0] = 1.0f; b[1] = 1.0f;
  v8f c = {};
  v8f d = __builtin_amdgcn_wmma_f32_16x16x4_f32(
      /*neg_a=*/false, a, /*neg_b=*/false, b,
      /*c_mod=*/(short)0, c, /*reuse_a=*/false, /*reuse_b=*/false);
  float s = ((d[0] + d[1]) + (d[2] + d[3])) + ((d[4] + d[5]) + (d[6] + d[7]));
  s += __shfl_xor(s, 16, 32);
  return s;  // every lane holds the full 32-lane sum
}

__device__ __forceinline__ void accum_elem(float lp, float lg, float w, int flat,
                                           float& s1, float& s2) {
  float sig  = 1.0f / (1.0f + __expf(-lg));   // sigmoid(logit)
  float r    = __logf(sig + EPS_F);           // log(sigmoid+eps)
  float tfac = (float)((flat & T_MASK) + 1);  // (t+1)
  s1 = fmaf(w * w * r * lp, tfac, s1);
  s2 += w;
}

__global__ void __launch_bounds__(NTHREADS)
rf_partial_kernel(const float4* __restrict__ lp4,
                  const float4* __restrict__ lg4,
                  const float4* __restrict__ w4,
                  float2* __restrict__ partial, int n4) {
  int tid    = blockIdx.x * NTHREADS + threadIdx.x;
  int stride = gridDim.x * NTHREADS;
  float s1 = 0.0f, s2 = 0.0f;
  // n4 = 2,097,152 and 2048*256 threads -> exactly 4 full b128-load iterations/thread.
  for (int i = tid; i < n4; i += stride) {
    float4 a = lp4[i];
    float4 g = lg4[i];
    float4 w = w4[i];
    int base = i << 2;
    accum_elem(a.x, g.x, w.x, base + 0, s1, s2);
    accum_elem(a.y, g.y, w.y, base + 1, s1, s2);
    accum_elem(a.z, g.z, w.z, base + 2, s1, s2);
    accum_elem(a.w, g.w, w.w, base + 3, s1, s2);
  }

  // Intra-wave reduction via the matrix unit (EXEC is all-1s here).
  float w1 = wave_sum_wmma(s1);
  float w2 = wave_sum_wmma(s2);

  __shared__ float l1[NWAVES], l2[NWAVES];
  int wave = threadIdx.x >> 5;
  if ((threadIdx.x & 31) == 0) { l1[wave] = w1; l2[wave] = w2; }
  __syncthreads();
  if (threadIdx.x == 0) {
    float t1 = 0.0f, t2 = 0.0f;
#pragma unroll
    for (int k = 0; k < NWAVES; ++k) { t1 += l1[k]; t2 += l2[k]; }
    partial[blockIdx.x] = make_float2(t1, t2);  // fully overwritten every launch
  }
}

__global__ void __launch_bounds__(NTHREADS)
rf_final_kernel(const float2* __restrict__ partial, float* __restrict__ out, int nb) {
  float s1 = 0.0f, s2 = 0.0f;
  for (int i = threadIdx.x; i < nb; i += NTHREADS) {  // nb = 2048 -> 8 pairs/thread
    float2 p = partial[i];
    s1 += p.x;
    s2 += p.y;
  }
  float w1 = wave_sum_wmma(s1);
  float w2 = wave_sum_wmma(s2);

  __shared__ float l1[NWAVES], l2[NWAVES];
  int wave = threadIdx.x >> 5;
  if ((threadIdx.x & 31) == 0) { l1[wave] = w1; l2[wave] = w2; }
  __syncthreads();
  if (threadIdx.x == 0) {
    float t1 = 0.0f, t2 = 0.0f;
#pragma unroll
    for (int k = 0; k < NWAVES; ++k) { t1 += l1[k]; t2 += l2[k]; }
    out[0] = t1 / t2;  // S1 / sum(weight)
  }
}

extern "C" void kernel_launch(void* const* d_in, const int* in_sizes, int n_in,
                              void* d_out, int out_size, void* d_ws, size_t ws_size,
                              hipStream_t stream) {
  // setup_inputs order: log_probs [B,T], logits [B,T,1], weight [B,T]; all float32.
  const float4* lp4 = (const float4*)d_in[0];
  const float4* lg4 = (const float4*)d_in[1];
  const float4* w4  = (const float4*)d_in[2];
  float* out = (float*)d_out;

  int n  = in_sizes[0];   // B*T = 8,388,608 (multiple of 4)
  int n4 = n >> 2;

  float2* partial = (float2*)d_ws;  // NBLOCKS * 8 bytes = 16 KB of scratch

  rf_partial_kernel<<<NBLOCKS, NTHREADS, 0, stream>>>(lp4, lg4, w4, partial, n4);
  rf_final_kernel<<<1, NTHREADS, 0, stream>>>(partial, out, NBLOCKS);
}